// PVRCNNPlusPlusBEVModule_730144440347
// MI455X (gfx1250) — compile-verified
//
#include <hip/hip_runtime.h>

// Problem constants (fixed by the reference's setup_inputs).
#define N_VOX    150000
#define CCH      128
#define NS       282752                         // D*H*W = 2*376*376
#define S_TILE   256                            // spatial slots per block (1/thread)
#define N_BLOCKS ((NS + S_TILE - 1) / S_TILE)   // 1105
#define MAP_PAD  (N_BLOCKS * S_TILE)            // 282880 (tile-padded map)

typedef float __attribute__((ext_vector_type(4))) v4f;

// ---------------------------------------------------------------------------
// K1: initialize the inverse map to -1 (empty). int4 stores.
// ---------------------------------------------------------------------------
__global__ __launch_bounds__(256) void map_init_kernel(int* __restrict__ map) {
    int i = blockIdx.x * blockDim.x + threadIdx.x;   // int4 granularity
    if (i < MAP_PAD / 4) {
        ((int4*)map)[i] = make_int4(-1, -1, -1, -1);
    }
}

// ---------------------------------------------------------------------------
// K2: scatter voxel ids into the dense map. atomicMax resolves duplicate
// flat_idx entries deterministically (largest voxel id wins), so graph
// replays re-validate identically.
// ---------------------------------------------------------------------------
__global__ __launch_bounds__(256) void map_scatter_kernel(
        const long long* __restrict__ flat_idx, int* __restrict__ map) {
    int v = blockIdx.x * blockDim.x + threadIdx.x;
    if (v < N_VOX) {
        int s = (int)flat_idx[v];
        atomicMax(&map[s], v);
    }
}

// ---------------------------------------------------------------------------
// K3: inverted gather, one thread per spatial slot.
//   - 1KB map tile staged global->LDS via the CDNA5 async DMA path (ASYNCcnt)
//   - reads : 32 x b128 gathers of the thread's 512B feature row
//             (full-line requests; features = 77MB, L2-resident on 192MB L2)
//   - writes: per-lane b32 NT stores, fully coalesced across the wave
//             (consecutive lanes own consecutive s) -> 145MB streamed once
// ---------------------------------------------------------------------------
__global__ __launch_bounds__(256) void bev_gather_kernel(
        const float* __restrict__ feat, const int* __restrict__ map,
        float* __restrict__ out) {
    __shared__ int smap[S_TILE];
    const int t  = threadIdx.x;
    const int s0 = blockIdx.x * S_TILE;

    // --- gfx1250 async copy: global -> LDS, 16B per lane, lanes 0..63 cover 1KB
    if (t < 64) {
        const int* g   = map + s0 + t * 4;
        unsigned   lds = (unsigned)(size_t)(&smap[t * 4]);  // low 32b = LDS offset
        asm volatile("global_load_async_to_lds_b128 %0, %1, off"
                     :: "v"(lds), "v"(g) : "memory");
#if __has_builtin(__builtin_amdgcn_s_wait_asynccnt)
        __builtin_amdgcn_s_wait_asynccnt(0);
#else
        asm volatile("s_wait_asynccnt 0" ::: "memory");
#endif
    }
    __syncthreads();

    const int s = s0 + t;
    if (s >= NS) return;

    const int  m     = smap[t];
    const bool valid = (m >= 0);
    // Clamped row pointer: loads are always in-bounds; emptiness handled by
    // a branchless vector select (v_cndmask), not divergent control flow.
    const v4f* frow = (const v4f*)(feat + (size_t)(valid ? m : 0) * CCH);
    const v4f  zero = {0.0f, 0.0f, 0.0f, 0.0f};

    float* o = out + s;
#pragma unroll 8
    for (int cc = 0; cc < CCH / 4; ++cc) {
        v4f r = frow[cc];          // one b128 gather: full 128B line utilization
        if (!valid) r = zero;      // cndmask x4
        const float r0 = r.x, r1 = r.y, r2 = r.z, r3 = r.w;
        // 4 NT b32 stores; each coalesces into one 128B line across the wave
        __builtin_nontemporal_store(r0, o + (size_t)(4 * cc + 0) * NS);
        __builtin_nontemporal_store(r1, o + (size_t)(4 * cc + 1) * NS);
        __builtin_nontemporal_store(r2, o + (size_t)(4 * cc + 2) * NS);
        __builtin_nontemporal_store(r3, o + (size_t)(4 * cc + 3) * NS);
    }
}

// ---------------------------------------------------------------------------
extern "C" void kernel_launch(void* const* d_in, const int* in_sizes, int n_in,
                              void* d_out, int out_size, void* d_ws, size_t ws_size,
                              hipStream_t stream) {
    (void)in_sizes; (void)n_in; (void)out_size; (void)ws_size;

    const float*     feat = (const float*)d_in[0];        // [150000,128] f32
    const long long* idx  = (const long long*)d_in[1];    // [150000] i64
    float*           out  = (float*)d_out;                // [1, C*D, H, W] f32
    int*             map  = (int*)d_ws;                   // MAP_PAD ints (~1.1MB)

    map_init_kernel   <<<(MAP_PAD / 4 + 255) / 256, 256, 0, stream>>>(map);
    map_scatter_kernel<<<(N_VOX + 255) / 256, 256, 0, stream>>>(idx, map);
    bev_gather_kernel <<<N_BLOCKS, 256, 0, stream>>>(feat, map, out);
}